// GramSchmidtLayer_58205396795619
// MI455X (gfx1250) — compile-verified
//
#include <hip/hip_runtime.h>
#include <hip/hip_bf16.h>

// Modified Gram-Schmidt, B=256 batches of complex [M=1024, R=64].
// Real-ified: per batch a 2048x64 f32 matrix, panel-blocked (16 cols/panel).
// Inter-panel projections run on V_WMMA_F32_16X16X4_F32 (full f32 precision);
// intra-panel MGS runs on VALU with wave32 reductions.
// Panel staging uses gfx1250 async global<->LDS (ASYNCcnt) b128 transfers.

#define EPS 1e-8f

typedef float v2f __attribute__((ext_vector_type(2)));
typedef float v8f __attribute__((ext_vector_type(8)));

constexpr int Mdim   = 1024;            // complex rows
constexpr int Rdim   = 64;              // columns
constexpr int PANEL  = 16;
constexpr int NPANEL = Rdim / PANEL;    // 4
constexpr int NROW   = 2 * Mdim;        // 2048 real rows
constexpr int LROWS  = 36;              // LDS stride (floats): 144 B rows -> 16B-aligned b128 granules,
                                        // conflict-free intra-panel walks, ~2/32 lane collisions in frag loads
constexpr int THREADS = 256;
constexpr int WAVES   = 8;
constexpr int ROWS_PER_WAVE = NROW / WAVES;   // 256

// LDS layout (float offsets)
constexpr int OFF_V    = 0;
constexpr int OFF_Q    = OFF_V + Mdim * LROWS;            // 36864
constexpr int CRED_RS  = 20;                              // padded row stride for partial-C
constexpr int CRED_WS  = 16 * CRED_RS;
constexpr int OFF_CRED = OFF_Q + Mdim * LROWS;            // 73728
constexpr int OFF_CMAT = OFF_CRED + WAVES * CRED_WS;      // 76288
constexpr int OFF_QN   = OFF_CMAT + 256;                  // 76544: ||q||^2 per column
constexpr int OFF_SC   = OFF_QN + Rdim;                   // 76608: reduction scratch
constexpr int LDS_FLOATS = OFF_SC + 16;                   // 76624 floats = 306,496 B <= 320 KB

// real-row n, column col -> LDS float offset within a panel buffer
__device__ __forceinline__ int vidx(int n, int col) {
  return (n >> 1) * LROWS + (col << 1) + (n & 1);
}

// ---- gfx1250 async global<->LDS transfers (tracked by ASYNCcnt) ----
__device__ __forceinline__ void async_g2l_b128(const void* sbase, int voff_bytes, unsigned lds_off_bytes) {
  // GLOBAL_LOAD_ASYNC_TO_LDS_B128, GVS mode: SGPR64 base + 32-bit per-lane offset
  asm volatile("global_load_async_to_lds_b128 %0, %1, %2"
               :
               : "v"(lds_off_bytes), "v"(voff_bytes), "s"(sbase)
               : "memory");
}
__device__ __forceinline__ void async_l2g_b128(void* sbase, int voff_bytes, unsigned lds_off_bytes) {
  // GLOBAL_STORE_ASYNC_FROM_LDS_B128
  asm volatile("global_store_async_from_lds_b128 %0, %1, %2"
               :
               : "v"(voff_bytes), "v"(lds_off_bytes), "s"(sbase)
               : "memory");
}
__device__ __forceinline__ void wait_async0() {
  asm volatile("s_wait_asynccnt 0" ::: "memory");
}

__global__ void __launch_bounds__(THREADS)
gram_schmidt_kernel(const float* __restrict__ in, float* __restrict__ out) {
  extern __shared__ float lds[];
  const int b    = blockIdx.x;
  const int tid  = threadIdx.x;
  const int wave = tid >> 5;
  const int lane = tid & 31;
  const int half = lane >> 4;    // 0: lanes 0-15, 1: lanes 16-31
  const int hl   = lane & 15;

  const size_t gbase = (size_t)b * (size_t)(Mdim * Rdim * 2);

  float* V  = lds + OFF_V;
  float* Q  = lds + OFF_Q;
  float* CR = lds + OFF_CRED;
  float* CM = lds + OFF_CMAT;
  float* QN = lds + OFF_QN;
  float* SC = lds + OFF_SC;

  for (int p = 0; p < NPANEL; ++p) {
    // ---- async-stage panel p of input into LDS V ----
    {
      const float* src = in + gbase + p * (PANEL * 2);
      for (int it = 0; it < 32; ++it) {
        int idx = tid + THREADS * it;   // 0..8191
        int m = idx >> 3, g = idx & 7;
        async_g2l_b128(src, m * 512 + g * 16,
                       (unsigned)((OFF_V + m * LROWS + g * 4) * 4));
      }
      wait_async0();
    }
    __syncthreads();

    // ---- project current panel against each finished panel q (block MGS, WMMA f32) ----
    for (int q = 0; q < p; ++q) {
      // async-stage Q panel q from the already-written output
      {
        const float* qsrc = out + gbase + q * (PANEL * 2);
        for (int it = 0; it < 32; ++it) {
          int idx = tid + THREADS * it;
          int m = idx >> 3, g = idx & 7;
          async_g2l_b128(qsrc, m * 512 + g * 16,
                         (unsigned)((OFF_Q + m * LROWS + g * 4) * 4));
        }
        wait_async0();
      }
      __syncthreads();

      // Gram: C[i][j] = sum_n Q[n][i] * V[n][j]; each wave accumulates its 256-row slice.
      v8f acc = {};
      const int nbase = wave * ROWS_PER_WAVE;
      for (int n0 = nbase; n0 < nbase + ROWS_PER_WAVE; n0 += 4) {
        const int nA = n0 + 2 * half;
        v2f a, bb;
        a.x  = Q[vidx(nA,     hl)];   // A[m=hl][k=0|2] = Q[n0+k][hl]
        a.y  = Q[vidx(nA + 1, hl)];   // A[m=hl][k=1|3]
        bb.x = V[vidx(nA,     hl)];   // B[k=0|2][n=hl] = V[n0+k][hl]
        bb.y = V[vidx(nA + 1, hl)];
        acc = __builtin_amdgcn_wmma_f32_16x16x4_f32(false, a, false, bb,
                                                    (short)0, acc, false, false);
      }
      #pragma unroll
      for (int v = 0; v < 8; ++v)     // D row = v + 8*half, col = hl
        CR[wave * CRED_WS + (v + 8 * half) * CRED_RS + hl] = acc[v];
      __syncthreads();

      // reduce the 8 partial C's, apply faithful denominator, pre-negate for update
      {
        const int i = tid >> 4, j = tid & 15;
        float s = 0.f;
        #pragma unroll
        for (int w = 0; w < WAVES; ++w) s += CR[w * CRED_WS + i * CRED_RS + j];
        CM[i * 16 + j] = -s / (QN[q * PANEL + i] + EPS);
      }
      __syncthreads();

      // update: V = Q * (-C) + V, tiled 16 rows per WMMA group
      for (int t = 0; t < ROWS_PER_WAVE / 16; ++t) {
        const int tb = wave * ROWS_PER_WAVE + t * 16;
        v8f accu;
        #pragma unroll
        for (int v = 0; v < 8; ++v)
          accu[v] = V[vidx(tb + v + 8 * half, hl)];
        #pragma unroll
        for (int i0 = 0; i0 < 16; i0 += 4) {
          v2f a, bb;
          a.x  = Q[vidx(tb + hl, i0 + 2 * half)];       // A[m=hl][k] = Q[tb+hl][i0+k]
          a.y  = Q[vidx(tb + hl, i0 + 2 * half + 1)];
          bb.x = CM[(i0 + 2 * half) * 16 + hl];         // B[k][n=hl] = -C[i0+k][hl]
          bb.y = CM[(i0 + 2 * half + 1) * 16 + hl];
          accu = __builtin_amdgcn_wmma_f32_16x16x4_f32(false, a, false, bb,
                                                       (short)0, accu, false, false);
        }
        #pragma unroll
        for (int v = 0; v < 8; ++v)
          V[vidx(tb + v + 8 * half, hl)] = accu[v];
      }
      __syncthreads();
    }

    // ---- within-panel sequential (modified) Gram-Schmidt on VALU ----
    for (int j = 0; j < PANEL; ++j) {
      for (int i = 0; i < j; ++i) {
        float part = 0.f;
        #pragma unroll
        for (int k = 0; k < 8; ++k) {
          const int n = tid + THREADS * k;
          part += V[vidx(n, j)] * V[vidx(n, i)];
        }
        #pragma unroll
        for (int off = 16; off > 0; off >>= 1) part += __shfl_xor(part, off, 32);
        if (lane == 0) SC[wave] = part;
        __syncthreads();
        if (tid == 0) {
          float s = 0.f;
          #pragma unroll
          for (int w = 0; w < WAVES; ++w) s += SC[w];
          SC[8] = s / (QN[p * PANEL + i] + EPS);
        }
        __syncthreads();
        const float coef = SC[8];
        #pragma unroll
        for (int k = 0; k < 8; ++k) {
          const int n = tid + THREADS * k;
          V[vidx(n, j)] -= coef * V[vidx(n, i)];
        }
        __syncthreads();
      }
      // normalize column j; record ||q||^2 = S/(S+eps) for faithful denominators
      {
        float part = 0.f;
        #pragma unroll
        for (int k = 0; k < 8; ++k) {
          const int n = tid + THREADS * k;
          const float x = V[vidx(n, j)];
          part += x * x;
        }
        #pragma unroll
        for (int off = 16; off > 0; off >>= 1) part += __shfl_xor(part, off, 32);
        if (lane == 0) SC[wave] = part;
        __syncthreads();
        if (tid == 0) {
          float s = 0.f;
          #pragma unroll
          for (int w = 0; w < WAVES; ++w) s += SC[w];
          const float denom = s + EPS;
          SC[8] = 1.0f / sqrtf(denom);
          QN[p * PANEL + j] = s / denom;
        }
        __syncthreads();
        const float rn = SC[8];
        #pragma unroll
        for (int k = 0; k < 8; ++k) {
          const int n = tid + THREADS * k;
          V[vidx(n, j)] *= rn;
        }
        __syncthreads();
      }
    }

    // ---- async write finished panel to output ----
    {
      float* dst = out + gbase + p * (PANEL * 2);
      for (int it = 0; it < 32; ++it) {
        int idx = tid + THREADS * it;
        int m = idx >> 3, g = idx & 7;
        async_l2g_b128(dst, m * 512 + g * 16,
                       (unsigned)((OFF_V + m * LROWS + g * 4) * 4));
      }
      wait_async0();
    }
    __threadfence();   // make Q panel visible to all waves for later re-read
    __syncthreads();
  }
}

extern "C" void kernel_launch(void* const* d_in, const int* in_sizes, int n_in,
                              void* d_out, int out_size, void* d_ws, size_t ws_size,
                              hipStream_t stream) {
  const float* in = (const float*)d_in[0];
  float* out = (float*)d_out;
  (void)in_sizes; (void)n_in; (void)out_size; (void)d_ws; (void)ws_size;

  const size_t shmem = (size_t)LDS_FLOATS * sizeof(float);   // ~300 KB of the 320 KB WGP LDS
  (void)hipFuncSetAttribute((const void*)gram_schmidt_kernel,
                            hipFuncAttributeMaxDynamicSharedMemorySize, (int)shmem);
  gram_schmidt_kernel<<<dim3(256), dim3(THREADS), shmem, stream>>>(in, out);
}